// CustomKeypointLoss_87608742904608
// MI455X (gfx1250) — compile-verified
//
#include <hip/hip_runtime.h>
#include <hip/hip_bf16.h>

// ---- problem constants (fixed by setup_inputs) ----
#define B_DIM 32
#define K_DIM 8
#define H_DIM 512
#define W_DIM 512
#define HW (H_DIM * W_DIM)          // 262144 floats per (b,k) map
#define NPAIR (B_DIM * K_DIM)       // 256
#define SEGS 8                      // segments per map -> 2048 blocks total
#define SEG_ELEMS (HW / SEGS)       // 32768 floats per block
#define THREADS 256
#define V4_PER_THREAD (SEG_ELEMS / 4 / THREADS) // 32 float4 loads per thread

typedef __attribute__((ext_vector_type(4))) float v4f;
typedef __attribute__((ext_vector_type(2))) float v2f;
typedef __attribute__((ext_vector_type(8))) float v8f;

struct PartMax { float v; int i; };

// ------------------------------------------------------------------
// Kernel 1: streaming argmax. One block = one (pair, segment).
// Non-temporal b128 loads (stream > L2), first-index-wins tie break.
// ------------------------------------------------------------------
__global__ __launch_bounds__(THREADS) void
argmax_part_kernel(const float* __restrict__ hm, PartMax* __restrict__ part) {
    const int seg  = blockIdx.x & (SEGS - 1);
    const int pair = blockIdx.x >> 3; // / SEGS
    const int t = threadIdx.x;

    const v4f* __restrict__ src =
        (const v4f*)(hm + (size_t)pair * HW + (size_t)seg * SEG_ELEMS);

    float bv = -__builtin_huge_valf();
    int   bi = 0x7fffffff;

    // ascending local index order => strict '>' keeps first occurrence
#pragma unroll 4
    for (int j = 0; j < V4_PER_THREAD; ++j) {
        const int o = t + j * THREADS;
        v4f v = __builtin_nontemporal_load(&src[o]);
        const int base = o * 4;
        if (v.x > bv) { bv = v.x; bi = base; }
        if (v.y > bv) { bv = v.y; bi = base + 1; }
        if (v.z > bv) { bv = v.z; bi = base + 2; }
        if (v.w > bv) { bv = v.w; bi = base + 3; }
    }

    __shared__ float sv[THREADS];
    __shared__ int   si[THREADS];
    sv[t] = bv;
    si[t] = bi + seg * SEG_ELEMS;   // global index within the map
    __syncthreads();

    for (int s = THREADS / 2; s > 0; s >>= 1) {
        if (t < s) {
            const float ov = sv[t + s];
            const int   oi = si[t + s];
            if (ov > sv[t] || (ov == sv[t] && oi < si[t])) {
                sv[t] = ov; si[t] = oi;
            }
        }
        __syncthreads();
    }
    if (t == 0) {
        PartMax p; p.v = sv[0]; p.i = si[0];
        part[blockIdx.x] = p;
    }
}

// ------------------------------------------------------------------
// Kernel 2: combine segment partials, compute loss terms, and reduce
// the 256 per-(b,k) contributions on the matrix unit:
//   D = ones(16x4) x B(4x16) + C  via V_WMMA_F32_16X16X4_F32
// With A == 1 the B lane layout is irrelevant for the total sum.
// ------------------------------------------------------------------
__global__ __launch_bounds__(THREADS) void
finish_kernel(const PartMax* __restrict__ part,
              const float* __restrict__ gt,
              const float* __restrict__ gmask,
              const float* __restrict__ nmask,
              const float* __restrict__ wmask,
              float* __restrict__ out) {
    const int t = threadIdx.x;
    const int b = t >> 3;            // pair -> batch
    // combine the SEGS partials with first-index tie break
    PartMax best = part[t * SEGS];
    for (int s = 1; s < SEGS; ++s) {
        PartMax p = part[t * SEGS + s];
        if (p.v > best.v || (p.v == best.v && p.i < best.i)) best = p;
    }
    const int idx = best.i;

    const float x = (float)(idx & (W_DIM - 1)) * (1.0f / (W_DIM - 1));
    const float y = (float)(idx >> 9)          * (1.0f / (H_DIM - 1));

    // reference: clip(x, 0, 511) is a no-op for x in [0,1]; then floor -> int
    const int ix = (int)floorf(fminf(fmaxf(x, 0.0f), 511.0f));
    const int iy = (int)floorf(fminf(fmaxf(y, 0.0f), 511.0f));
    const float quant = fabsf(x - (float)ix) + fabsf(y - (float)iy);

    const size_t mo = (size_t)b * HW + (size_t)ix * W_DIM + (size_t)iy;
    const float gm = gmask[mo];
    const float nm = nmask[mo];
    const float wm = wmask[mo];

    const float gtx = gt[t * 2 + 0];
    const float gty = gt[t * 2 + 1];

    // per-(b,k): |pk-gt|_1 + 10*quant*3 + 10*10*((1-g)+(1-n)+(1-w))
    const float contrib =
        fabsf(x - gtx) + fabsf(y - gty) +
        30.0f * quant +
        100.0f * ((1.0f - gm) + (1.0f - nm) + (1.0f - wm));

    __shared__ float sh[THREADS];
    sh[t] = contrib;
    __syncthreads();

    if (t < 32) {                    // wave 0 only; EXEC all-1s for this wave
        v8f c = {};
        v2f a; a.x = 1.0f; a.y = 1.0f;   // all-ones A (16x4)
#pragma unroll
        for (int i = 0; i < 4; ++i) {
            v2f bm;
            bm.x = sh[i * 64 + t];
            bm.y = sh[i * 64 + 32 + t];  // 64 values per WMMA, 4 WMMAs = 256
            c = __builtin_amdgcn_wmma_f32_16x16x4_f32(
                    false, a, false, bm, (short)0, c, false, false);
        }
        // c[0]: lanes 0-15 = D[0][n] (column sums), lanes 16-31 = D[8][n] (same)
        float v = c[0];
        for (int off = 16; off > 0; off >>= 1)
            v += __shfl_xor(v, off, 32);
        if (t == 0) out[0] = 0.5f * v;   // 32 lanes hold 2x the total
    }
}

extern "C" void kernel_launch(void* const* d_in, const int* in_sizes, int n_in,
                              void* d_out, int out_size, void* d_ws, size_t ws_size,
                              hipStream_t stream) {
    const float* hm = (const float*)d_in[0];  // (32,8,512,512)
    const float* gt = (const float*)d_in[1];  // (32,8,2)
    const float* gm = (const float*)d_in[2];  // (32,512,512)
    const float* nm = (const float*)d_in[3];
    const float* wm = (const float*)d_in[4];

    PartMax* part = (PartMax*)d_ws;           // 2048 * 8B = 16 KiB scratch

    argmax_part_kernel<<<NPAIR * SEGS, THREADS, 0, stream>>>(hm, part);
    finish_kernel<<<1, THREADS, 0, stream>>>(part, gt, gm, nm, wm, (float*)d_out);
}